// TextualEntailmentModel_13675175871137
// MI455X (gfx1250) — compile-verified
//
#include <hip/hip_runtime.h>
#include <hip/hip_bf16.h>
#include <math.h>
#include <stdint.h>

typedef float v2f __attribute__((ext_vector_type(2)));
typedef float v8f __attribute__((ext_vector_type(8)));

#define LSEQ   256
#define BSZ    128
#define EDIM   1024
#define HDIM   1024
#define UDIM   4096
#define NCLS   3

#define KCHUNK 32
#define A_LDS_STRIDE 36   // 32 + 4 pad floats: 144B rows, 16B aligned, bank-spread

// ---------------- workspace layout (bytes) ----------------
#define WS_TOKF   0                       // int[2*128]
#define WS_TSF    1024                    // float[2*128]
#define WS_FHID   2048                    // float[2*128*1024]
#define WS_U      (2048 + 2*128*1024*4)   // float[128*4096]
#define WS_H      (WS_U + 128*4096*4)     // float[128*1024]
#define WS_STATS  (WS_H + 128*1024*4)     // float[2*128*3]

// ---------------------------------------------------------------------------
// CDNA5 async global->LDS copy (ASYNCcnt-tracked), 16B per lane.
// dsaddr = LDS byte address (VGPR), v[addr:addr+1] = global address.
// ---------------------------------------------------------------------------
__device__ __forceinline__ void async_copy_b128(uint32_t lds_off, const void* gptr) {
    asm volatile("global_load_async_to_lds_b128 %0, %1, off"
                 :: "v"(lds_off), "v"(gptr)
                 : "memory");
}
__device__ __forceinline__ void wait_asynccnt0() {
    asm volatile("s_wait_asynccnt 0x0" ::: "memory");
}
__device__ __forceinline__ uint32_t lds_addr32(const void* p) {
    // generic pointer to LDS: low 32 bits are the LDS byte offset
    return (uint32_t)(uintptr_t)p;
}

// ---------------------------------------------------------------------------
// 1) per-(seq,batch): sequence length, Ts, final-token id
// ---------------------------------------------------------------------------
__global__ void lens_kernel(const int* __restrict__ seq1,
                            const int* __restrict__ seq2,
                            int* __restrict__ tokf,
                            float* __restrict__ Tsf) {
    const int s = blockIdx.x;          // 0..1
    const int b = threadIdx.x;         // 0..127
    const int* sq = s ? seq2 : seq1;
    int cnt = 0;
    for (int t = 0; t < LSEQ; ++t) cnt += (sq[t * BSZ + b] == 0);
    const int lens = LSEQ - cnt - 1;
    const int Ts   = 2 * lens - 1;
    const int tf   = Ts - 1;                   // index into seq_ext
    const int trow = (tf < LSEQ) ? tf : tf - LSEQ;
    tokf[s * BSZ + b] = sq[trow * BSZ + b];
    Tsf[s * BSZ + b]  = (float)Ts;
}

// ---------------------------------------------------------------------------
// 2) fhid = tanh(gather(emb) @ W_top + b_top)
//    WMMA f32 16x16x4, async-LDS staged tiles.
//    grid(16, 8, 2), block 128 (4 waves). Block tile: M=16, N=64, Kchunk=32.
// ---------------------------------------------------------------------------
__global__ void fhid_wmma(const float* __restrict__ emb,
                          const float* __restrict__ W_top,
                          const float* __restrict__ b_top,
                          const int*   __restrict__ tokf,
                          float* __restrict__ fhid) {
    __shared__ __align__(16) float As[16 * A_LDS_STRIDE];   // 2.25 KB
    __shared__ __align__(16) float Bs[KCHUNK * 64];         // 8 KB

    const int tid  = threadIdx.x;
    const int lane = tid & 31;
    const int wave = tid >> 5;
    const int s    = blockIdx.z;
    const int m0   = blockIdx.y * 16;
    const int nb   = blockIdx.x * 64;          // block's first output column
    const int l15  = lane & 15;
    const int kh   = (lane >> 4) * 2;          // 0 or 2: K sub-pair per half-wave
    const int ncol = nb + wave * 16 + l15;     // this lane's output column

    // staging assignments
    const int ar  = tid >> 3;                  // A row 0..15
    const int ac4 = (tid & 7) * 4;             // A col chunk (floats)
    const float* arow_g = emb + (size_t)tokf[s * BSZ + m0 + ar] * EDIM;
    const uint32_t a_dst = lds_addr32(&As[ar * A_LDS_STRIDE + ac4]);

    v8f acc = {0.f,0.f,0.f,0.f,0.f,0.f,0.f,0.f};
    for (int k = 0; k < EDIM; k += KCHUNK) {
        // stage A tile: 16 x 32 f32, one b128 per thread
        async_copy_b128(a_dst, arow_g + k + ac4);
        // stage B tile: 32 x 64 f32, four b128 per thread
#pragma unroll
        for (int j = 0; j < 4; ++j) {
            const int idx = tid + j * 128;     // 0..511
            const int br  = idx >> 4;          // K row 0..31
            const int bc4 = (idx & 15) * 4;    // col chunk
            async_copy_b128(lds_addr32(&Bs[br * 64 + bc4]),
                            W_top + (size_t)(k + br) * HDIM + nb + bc4);
        }
        wait_asynccnt0();
        __syncthreads();

#pragma unroll
        for (int kk = 0; kk < KCHUNK; kk += 4) {
            v2f a, bb;
            a.x  = As[l15 * A_LDS_STRIDE + kk + kh];
            a.y  = As[l15 * A_LDS_STRIDE + kk + kh + 1];
            bb.x = Bs[(kk + kh    ) * 64 + wave * 16 + l15];
            bb.y = Bs[(kk + kh + 1) * 64 + wave * 16 + l15];
            acc  = __builtin_amdgcn_wmma_f32_16x16x4_f32(false, a, false, bb,
                                                         (short)0, acc, false, false);
        }
        __syncthreads();   // before next chunk overwrites tiles
    }

    const float bias = b_top[ncol];
    float* C = fhid + (size_t)s * BSZ * HDIM;
    const int rbase = m0 + ((lane >> 4) * 8);
#pragma unroll
    for (int j = 0; j < 8; ++j)
        C[(size_t)(rbase + j) * HDIM + ncol] = tanhf(acc[j] + bias);
}

// ---------------------------------------------------------------------------
// 3) u = [f1, f2, |f1-f2|, f1*f2]
// ---------------------------------------------------------------------------
__global__ void u_kernel(const float* __restrict__ fhid, float* __restrict__ u) {
    const int i = blockIdx.x * blockDim.x + threadIdx.x;   // 128*1024 threads
    const int b = i >> 10, h = i & 1023;
    const float f1 = fhid[(size_t)b * HDIM + h];
    const float f2 = fhid[(size_t)(BSZ + b) * HDIM + h];
    float* ur = u + (size_t)b * UDIM;
    ur[h]            = f1;
    ur[HDIM + h]     = f2;
    ur[2*HDIM + h]   = fabsf(f1 - f2);
    ur[3*HDIM + h]   = f1 * f2;
}

// ---------------------------------------------------------------------------
// 4) h = relu(u @ W1 + b1)   M=128, N=1024, K=4096
//    WMMA f32 16x16x4, async-LDS staged tiles. grid(16, 8), block 128.
// ---------------------------------------------------------------------------
__global__ void clf_wmma(const float* __restrict__ u,
                         const float* __restrict__ W1,
                         const float* __restrict__ b1,
                         float* __restrict__ h) {
    __shared__ __align__(16) float As[16 * A_LDS_STRIDE];
    __shared__ __align__(16) float Bs[KCHUNK * 64];

    const int tid  = threadIdx.x;
    const int lane = tid & 31;
    const int wave = tid >> 5;
    const int m0   = blockIdx.y * 16;
    const int nb   = blockIdx.x * 64;
    const int l15  = lane & 15;
    const int kh   = (lane >> 4) * 2;
    const int ncol = nb + wave * 16 + l15;

    const int ar  = tid >> 3;
    const int ac4 = (tid & 7) * 4;
    const float* arow_g = u + (size_t)(m0 + ar) * UDIM;
    const uint32_t a_dst = lds_addr32(&As[ar * A_LDS_STRIDE + ac4]);

    v8f acc = {0.f,0.f,0.f,0.f,0.f,0.f,0.f,0.f};
    for (int k = 0; k < UDIM; k += KCHUNK) {
        async_copy_b128(a_dst, arow_g + k + ac4);
#pragma unroll
        for (int j = 0; j < 4; ++j) {
            const int idx = tid + j * 128;
            const int br  = idx >> 4;
            const int bc4 = (idx & 15) * 4;
            async_copy_b128(lds_addr32(&Bs[br * 64 + bc4]),
                            W1 + (size_t)(k + br) * HDIM + nb + bc4);
        }
        wait_asynccnt0();
        __syncthreads();

#pragma unroll
        for (int kk = 0; kk < KCHUNK; kk += 4) {
            v2f a, bb;
            a.x  = As[l15 * A_LDS_STRIDE + kk + kh];
            a.y  = As[l15 * A_LDS_STRIDE + kk + kh + 1];
            bb.x = Bs[(kk + kh    ) * 64 + wave * 16 + l15];
            bb.y = Bs[(kk + kh + 1) * 64 + wave * 16 + l15];
            acc  = __builtin_amdgcn_wmma_f32_16x16x4_f32(false, a, false, bb,
                                                         (short)0, acc, false, false);
        }
        __syncthreads();
    }

    const float bias = b1[ncol];
    const int rbase = m0 + ((lane >> 4) * 8);
#pragma unroll
    for (int j = 0; j < 8; ++j) {
        const float v = acc[j] + bias;
        h[(size_t)(rbase + j) * HDIM + ncol] = v > 0.f ? v : 0.f;
    }
}

// ---------------------------------------------------------------------------
// 5) res_clf = h @ W2 + b2   (128 x 3)
// ---------------------------------------------------------------------------
__global__ void out_kernel(const float* __restrict__ h,
                           const float* __restrict__ W2,
                           const float* __restrict__ b2,
                           float* __restrict__ out) {
    const int i = threadIdx.x;                 // 384 threads
    const int b = i / NCLS, c = i % NCLS;
    float acc = b2[c];
    const float* hr = h + (size_t)b * HDIM;
    for (int k = 0; k < HDIM; ++k) acc += hr[k] * W2[k * NCLS + c];
    out[b * NCLS + c] = acc;
}

// ---------------------------------------------------------------------------
// 6) ragged softmax stats. One block per (b, s); 8 waves each own timesteps.
//    seq_ext row t>=256 duplicates row t-256 -> weight = [t<Ts] + [t+256<Ts].
// ---------------------------------------------------------------------------
__global__ void act_stats(const int* __restrict__ seq1,
                          const int* __restrict__ seq2,
                          const float* __restrict__ emb,
                          const float* __restrict__ W_act,
                          const float* __restrict__ b_act,
                          const float* __restrict__ Tsf,
                          float* __restrict__ stats) {
    const int b = blockIdx.x, s = blockIdx.y;
    const int* sq = s ? seq2 : seq1;
    const float Ts = Tsf[s * BSZ + b];
    const int lane = threadIdx.x & 31, wave = threadIdx.x >> 5;
    const float ba0 = b_act[0], ba1 = b_act[1];

    float s_push = 0.f, s_pop = 0.f, s_sq = 0.f;
    for (int t = wave; t < LSEQ; t += 8) {
        const int tok = sq[t * BSZ + b];
        const float* e = emb + (size_t)tok * EDIM;
        float a0 = 0.f, a1 = 0.f;
        for (int i = lane; i < EDIM; i += 32) {
            const float ev = e[i];
            a0 += ev * W_act[i * 2 + 0];
            a1 += ev * W_act[i * 2 + 1];
        }
        for (int off = 16; off; off >>= 1) {     // butterfly: all lanes get sum
            a0 += __shfl_xor(a0, off, 32);
            a1 += __shfl_xor(a1, off, 32);
        }
        a0 += ba0; a1 += ba1;
        const float m  = fmaxf(a0, a1);
        const float e0 = __expf(a0 - m), e1 = __expf(a1 - m);
        const float inv = 1.f / (e0 + e1);
        const float p0 = e0 * inv, p1 = e1 * inv;          // push, pop
        const float w = ((float)t < Ts ? 1.f : 0.f) +
                        ((float)(t + LSEQ) < Ts ? 1.f : 0.f);
        s_push += w * p0;
        s_pop  += w * p1;
        s_sq   += w * (p0 - p1) * (p0 - p1);
    }
    __shared__ float red[8][3];
    if (lane == 0) { red[wave][0] = s_push; red[wave][1] = s_pop; red[wave][2] = s_sq; }
    __syncthreads();
    if (threadIdx.x == 0) {
        float P = 0.f, Q = 0.f, S = 0.f;
        for (int wv = 0; wv < 8; ++wv) { P += red[wv][0]; Q += red[wv][1]; S += red[wv][2]; }
        float* st = stats + (size_t)(s * BSZ + b) * 3;
        st[0] = P; st[1] = Q; st[2] = S;
    }
}

// ---------------------------------------------------------------------------
// 7) finalize: dis / diff scalars via deterministic tree reduction
// ---------------------------------------------------------------------------
__global__ void finalize_kernel(const float* __restrict__ stats,
                                const float* __restrict__ Tsf,
                                float* __restrict__ out) {
    __shared__ float A[4][BSZ];
    const int b = threadIdx.x;
    for (int s = 0; s < 2; ++s) {
        const float Tf = Tsf[s * BSZ + b];
        const float* st = stats + (size_t)(s * BSZ + b) * 3;
        const float sum_push = st[0] / Tf;
        const float sum_pop  = (st[1] + 1.f) / Tf;
        const float d = sum_push - sum_pop;
        A[s][b]     = d * d;               // for dis_s
        A[2 + s][b] = sqrtf(st[2]) / Tf;   // for diff_s
    }
    __syncthreads();
    for (int stp = BSZ / 2; stp; stp >>= 1) {
        if (b < stp)
            for (int j = 0; j < 4; ++j) A[j][b] += A[j][b + stp];
        __syncthreads();
    }
    if (b == 0) {
        out[BSZ * NCLS + 0] = sqrtf(A[0][0]) / (float)BSZ;  // dis1
        out[BSZ * NCLS + 1] = sqrtf(A[1][0]) / (float)BSZ;  // dis2
        out[BSZ * NCLS + 2] = A[2][0] / (float)BSZ;         // diff1
        out[BSZ * NCLS + 3] = A[3][0] / (float)BSZ;         // diff2
    }
}

// ---------------------------------------------------------------------------
extern "C" void kernel_launch(void* const* d_in, const int* in_sizes, int n_in,
                              void* d_out, int out_size, void* d_ws, size_t ws_size,
                              hipStream_t stream) {
    const int*   seq1  = (const int*)  d_in[0];
    const int*   seq2  = (const int*)  d_in[1];
    const float* emb   = (const float*)d_in[2];
    const float* W_top = (const float*)d_in[3];
    const float* b_top = (const float*)d_in[4];
    const float* W_act = (const float*)d_in[5];
    const float* b_act = (const float*)d_in[6];
    const float* W1    = (const float*)d_in[7];
    const float* b1    = (const float*)d_in[8];
    const float* W2    = (const float*)d_in[9];
    const float* b2    = (const float*)d_in[10];
    float* out = (float*)d_out;

    char* ws = (char*)d_ws;
    int*   tokf  = (int*)  (ws + WS_TOKF);
    float* Tsf   = (float*)(ws + WS_TSF);
    float* fhid  = (float*)(ws + WS_FHID);
    float* u     = (float*)(ws + WS_U);
    float* h     = (float*)(ws + WS_H);
    float* stats = (float*)(ws + WS_STATS);

    lens_kernel<<<dim3(2), dim3(BSZ), 0, stream>>>(seq1, seq2, tokf, Tsf);

    fhid_wmma<<<dim3(HDIM / 64, BSZ / 16, 2), dim3(128), 0, stream>>>(
        emb, W_top, b_top, tokf, fhid);

    u_kernel<<<dim3((BSZ * HDIM) / 256), dim3(256), 0, stream>>>(fhid, u);

    clf_wmma<<<dim3(HDIM / 64, BSZ / 16, 1), dim3(128), 0, stream>>>(u, W1, b1, h);

    out_kernel<<<dim3(1), dim3(BSZ * NCLS), 0, stream>>>(h, W2, b2, out);

    act_stats<<<dim3(BSZ, 2), dim3(256), 0, stream>>>(
        seq1, seq2, emb, W_act, b_act, Tsf, stats);

    finalize_kernel<<<dim3(1), dim3(BSZ), 0, stream>>>(stats, Tsf, out);
}